// Model_text_sp_28003186770674
// MI455X (gfx1250) — compile-verified
//
#include <hip/hip_runtime.h>
#include <hip/hip_bf16.h>

typedef float v2f __attribute__((ext_vector_type(2)));
typedef float v8f __attribute__((ext_vector_type(8)));
typedef float v4f __attribute__((ext_vector_type(4)));

#define LATDIM 64
#define TXTDIM 768

// ---------------------------------------------------------------------------
// Dense text projection + bias + row-wise L2 normalize, using fp32 WMMA.
// One wave (32 lanes) per 16-row tile. D = A(16x4) x B(4x16) + C, looped over
// K=768 in steps of 4, with 4 independent accumulators covering N=0..63.
//
// VGPR layouts assumed per CDNA5 ISA 7.12.2:
//   A 16x4 f32 : VGPR j -> K = j + 2*(lane>=16), M = lane&15  (contig pair -> b64)
//   B 4x16 f32 : VGPR j -> K = j + 2*(lane>=16), N = lane&15
//   C/D 16x16  : VGPR v -> M = v + 8*(lane>=16), N = lane&15
// ---------------------------------------------------------------------------
__global__ void __launch_bounds__(32)
text_proj_norm_kernel(const float* __restrict__ text,   // [item, 768]
                      const float* __restrict__ W,      // [768, 64]
                      const float* __restrict__ bias,   // [64]
                      float* __restrict__ out,          // [item, 64] normalized
                      int item)
{
    const int lane  = threadIdx.x & 31;
    const int m     = lane & 15;
    const int khalf = lane >> 4;            // 0 or 1
    const int row0  = blockIdx.x * 16;

    int arow = row0 + m;
    if (arow >= item) arow = item - 1;      // clamp (item is a multiple of 16 anyway)
    const float* aptr = text + (size_t)arow * TXTDIM + 2 * khalf;

    v8f c0 = {}, c1 = {}, c2 = {}, c3 = {};

    for (int k0 = 0; k0 < TXTDIM; k0 += 4) {
        v2f a = *(const v2f*)(aptr + k0);

        v2f b0, b1, b2, b3;
#pragma unroll
        for (int j = 0; j < 2; ++j) {
            const int k = k0 + j + 2 * khalf;
            const float* wrow = W + (size_t)k * LATDIM + m;
            b0[j] = wrow[0];
            b1[j] = wrow[16];
            b2[j] = wrow[32];
            b3[j] = wrow[48];
        }
        c0 = __builtin_amdgcn_wmma_f32_16x16x4_f32(false, a, false, b0, (short)0, c0, false, false);
        c1 = __builtin_amdgcn_wmma_f32_16x16x4_f32(false, a, false, b1, (short)0, c1, false, false);
        c2 = __builtin_amdgcn_wmma_f32_16x16x4_f32(false, a, false, b2, (short)0, c2, false, false);
        c3 = __builtin_amdgcn_wmma_f32_16x16x4_f32(false, a, false, b3, (short)0, c3, false, false);
    }

    // bias depends only on column n = ntile*16 + m
    const float bi0 = bias[m];
    const float bi1 = bias[16 + m];
    const float bi2 = bias[32 + m];
    const float bi3 = bias[48 + m];
#pragma unroll
    for (int v = 0; v < 8; ++v) {
        c0[v] += bi0; c1[v] += bi1; c2[v] += bi2; c3[v] += bi3;
    }

    // Row-wise L2 normalize. VGPR v holds row (v + 8*khalf); reduce squares
    // across the 16 lanes of this half-wave (xor masks < 16 stay in-half).
#pragma unroll
    for (int v = 0; v < 8; ++v) {
        float ss = c0[v]*c0[v] + c1[v]*c1[v] + c2[v]*c2[v] + c3[v]*c3[v];
        ss += __shfl_xor(ss, 1, 32);
        ss += __shfl_xor(ss, 2, 32);
        ss += __shfl_xor(ss, 4, 32);
        ss += __shfl_xor(ss, 8, 32);
        const float inv = 1.0f / fmaxf(sqrtf(ss), 1e-12f);

        const int row = row0 + v + 8 * khalf;
        if (row < item) {
            float* o = out + (size_t)row * LATDIM + m;
            o[0]  = c0[v] * inv;
            o[16] = c1[v] * inv;
            o[32] = c2[v] * inv;
            o[48] = c3[v] * inv;
        }
    }
}

// ---------------------------------------------------------------------------
// COO SpMM with scatter atomics: out[rows[e]] += scale*vals[e] * x[cols[e]]
// Input x is the virtual concat(srcU [userCount rows], srcI [rest]).
// One wave per edge; lane handles features {2*lane, 2*lane+1}. Edge metadata
// is loaded 32-at-a-time per lane and broadcast with shuffles.
// ---------------------------------------------------------------------------
__global__ void __launch_bounds__(256)
spmm_atomic_kernel(const int* __restrict__ rows, const int* __restrict__ cols,
                   const float* __restrict__ vals,
                   const float* __restrict__ srcU, const float* __restrict__ srcI,
                   float* __restrict__ out, float scale, int nEdges, int userCount)
{
    const int lane   = threadIdx.x & 31;
    const int wave   = blockIdx.x * (blockDim.x >> 5) + (threadIdx.x >> 5);
    const int nWaves = gridDim.x * (blockDim.x >> 5);

    for (int base = wave * 32; base < nEdges; base += nWaves * 32) {
        const int myE = base + lane;
        int   r_l = 0, c_l = 0;
        float v_l = 0.0f;
        if (myE < nEdges) {
            r_l = rows[myE];
            c_l = cols[myE];
            v_l = vals[myE] * scale;
        }
        const int rem = nEdges - base;
        const int cnt = rem < 32 ? rem : 32;
        for (int j = 0; j < cnt; ++j) {
            const int   r = __shfl(r_l, j, 32);
            const int   c = __shfl(c_l, j, 32);
            const float v = __shfl(v_l, j, 32);
            const float* x = (c < userCount) ? (srcU + (size_t)c * LATDIM)
                                             : (srcI + (size_t)(c - userCount) * LATDIM);
            const v2f xv = *(const v2f*)(x + 2 * lane);
            float* o = out + (size_t)r * LATDIM + 2 * lane;
            atomicAdd(o,     v * xv[0]);
            atomicAdd(o + 1, v * xv[1]);
        }
    }
}

// ---------------------------------------------------------------------------
// Elementwise helpers (float4-vectorized, grid-stride)
// ---------------------------------------------------------------------------
__global__ void __launch_bounds__(256)
zero4_kernel(v4f* __restrict__ p, int n4)
{
    for (int i = blockIdx.x * blockDim.x + threadIdx.x; i < n4; i += gridDim.x * blockDim.x)
        p[i] = (v4f){0.f, 0.f, 0.f, 0.f};
}

// b += a; out = b   (finalize embeds and seed the running total)
__global__ void __launch_bounds__(256)
add_store4_kernel(v4f* __restrict__ b, const v4f* __restrict__ a,
                  v4f* __restrict__ outT, int n4)
{
    for (int i = blockIdx.x * blockDim.x + threadIdx.x; i < n4; i += gridDim.x * blockDim.x) {
        const v4f s = b[i] + a[i];
        b[i] = s;
        outT[i] = s;
    }
}

// out += a
__global__ void __launch_bounds__(256)
accum4_kernel(v4f* __restrict__ outT, const v4f* __restrict__ a, int n4)
{
    for (int i = blockIdx.x * blockDim.x + threadIdx.x; i < n4; i += gridDim.x * blockDim.x)
        outT[i] += a[i];
}

// ---------------------------------------------------------------------------
extern "C" void kernel_launch(void* const* d_in, const int* in_sizes, int n_in,
                              void* d_out, int out_size, void* d_ws, size_t ws_size,
                              hipStream_t stream)
{
    const float* uEmb = (const float*)d_in[0];
    const float* iEmb = (const float*)d_in[1];
    const float* text = (const float*)d_in[2];
    const float* Wt   = (const float*)d_in[3];
    const float* bt   = (const float*)d_in[4];
    const float* aV   = (const float*)d_in[5];
    const float* tV   = (const float*)d_in[6];
    const int*   aR   = (const int*)d_in[7];
    const int*   aC   = (const int*)d_in[8];
    const int*   tR   = (const int*)d_in[9];
    const int*   tC   = (const int*)d_in[10];
    float* outT = (float*)d_out;

    const int user = in_sizes[0] / LATDIM;       // 100000
    const int item = in_sizes[1] / LATDIM;       // 50000
    const int nTot = user + item;                // 150000
    const int E    = in_sizes[5];                // 2400000

    // scratch layout: tfn [item*64] | A [nTot*64] | B [nTot*64]
    float* tfn = (float*)d_ws;
    float* A   = tfn + (size_t)item * LATDIM;
    float* B   = A   + (size_t)nTot * LATDIM;

    const int NF  = nTot * LATDIM;               // 9,600,000 floats
    const int NF4 = NF / 4;
    const int ewGrid = (NF4 + 255) / 256;
    const int spGrid = 2048;                     // 16384 waves for 2.4M edges

    // zero accumulators
    zero4_kernel<<<ewGrid, 256, 0, stream>>>((v4f*)A, NF4);
    zero4_kernel<<<ewGrid, 256, 0, stream>>>((v4f*)B, NF4);

    // tfn = l2norm(text @ W + b)           (WMMA f32 16x16x4)
    text_proj_norm_kernel<<<(item + 15) / 16, 32, 0, stream>>>(text, Wt, bt, tfn, item);

    // B = 0.2 * spmm(T, concat(u, i))              (= RIS_LAMBDA * embeds_ta)
    spmm_atomic_kernel<<<spGrid, 256, 0, stream>>>(tR, tC, tV, uEmb, iEmb, B, 0.2f, E, user);

    // A = spmm(adj, concat(u, tfn))                (= e1)
    spmm_atomic_kernel<<<spGrid, 256, 0, stream>>>(aR, aC, aV, uEmb, tfn, A, 1.0f, E, user);

    // B += spmm(adj, concat(e1[:user], i))         (= 0.2*ta + e2)
    spmm_atomic_kernel<<<spGrid, 256, 0, stream>>>(aR, aC, aV, A, iEmb, B, 1.0f, E, user);

    // B = e1 + e2 + 0.2*ta = embeds ; outT = embeds (layer-0 contribution)
    add_store4_kernel<<<ewGrid, 256, 0, stream>>>((v4f*)B, (const v4f*)A, (v4f*)outT, NF4);

    // GNN layer 1: A = spmm(adj, embeds); outT += A
    zero4_kernel<<<ewGrid, 256, 0, stream>>>((v4f*)A, NF4);
    spmm_atomic_kernel<<<spGrid, 256, 0, stream>>>(aR, aC, aV, B, B + (size_t)user * LATDIM,
                                                   A, 1.0f, E, user);
    accum4_kernel<<<ewGrid, 256, 0, stream>>>((v4f*)outT, (const v4f*)A, NF4);

    // GNN layer 2: B = spmm(adj, A); outT += B
    zero4_kernel<<<ewGrid, 256, 0, stream>>>((v4f*)B, NF4);
    spmm_atomic_kernel<<<spGrid, 256, 0, stream>>>(aR, aC, aV, A, A + (size_t)user * LATDIM,
                                                   B, 1.0f, E, user);
    accum4_kernel<<<ewGrid, 256, 0, stream>>>((v4f*)outT, (const v4f*)B, NF4);
}